// EncoderI_75256416961015
// MI455X (gfx1250) — compile-verified
//
#include <hip/hip_runtime.h>

// ---------------------------------------------------------------------------
// GATv2, 3 layers, N=50000 nodes, E=600000 edges, IN=128, H=4, C=32.
// GEMMs via V_WMMA_F32_16X16X4_F32 (full f32 precision), register-blocked so
// one wave computes a 16-row x M-col strip (A fragment reused across CT=M/16
// WMMAs).  Edge passes use native f32 atomics (ordered-uint atomicMax for
// segment_max, atomic_add_f32 for the weighted scatter).
// ---------------------------------------------------------------------------

typedef __attribute__((ext_vector_type(2))) float v2f;
typedef __attribute__((ext_vector_type(8))) float v8f;

#define GAT_N 50000
#define GAT_E 600000
#define GAT_H 4
#define GAT_C 32
#define GAT_F 128   // H*C

// Order-preserving float<->uint mapping for atomicMax-based segment_max.
__device__ __forceinline__ unsigned fkey(float f) {
    int i = __float_as_int(f);
    return (i >= 0) ? ((unsigned)i | 0x80000000u) : ~(unsigned)i;
}
__device__ __forceinline__ float funkey(unsigned k) {
    int i = (k & 0x80000000u) ? (int)(k & 0x7fffffffu) : (int)(~k);
    return __int_as_float(i);
}

// ---------------------------------------------------------------------------
// Y[Nrows x M] = X[Nrows x K] @ W[K x M] + bias[M],  M == CT*16.
// One wave32 per 16-row strip; CT accumulators cover all M columns, so each
// A fragment (16x4 f32) feeds CT WMMAs.  Requires Nrows%16==0, K%4==0.
// A layout (ISA 7.12.2, 32-bit A 16x4): lanes 0-15 = rows, VGPR0/1 = K,K+1;
// lanes 16-31 carry K+2,K+3.  B mirrors with N striped across lanes.
// C/D layout: VGPR r -> row r (lanes 0-15) / row r+8 (lanes 16-31), col=lane&15.
// ---------------------------------------------------------------------------
template <int CT>
__global__ __launch_bounds__(128) void gemm_wmma_strip(
    const float* __restrict__ X, const float* __restrict__ W,
    const float* __restrict__ bias, float* __restrict__ Y,
    int Nrows, int K) {
    constexpr int M = CT * 16;
    const int lane = threadIdx.x & 31;
    const int wave = threadIdx.x >> 5;
    const int rt = blockIdx.x * (blockDim.x >> 5) + wave;
    if (rt >= (Nrows >> 4)) return;       // wave-uniform: EXEC stays all-ones
    const int row0 = rt << 4;
    const int half = lane >> 4;           // 0: K pair {k,k+1}, 1: {k+2,k+3}
    const int l16  = lane & 15;

    v8f acc[CT];
#pragma unroll
    for (int t = 0; t < CT; ++t) acc[t] = (v8f){};

    const float* __restrict__ xrow = X + (size_t)(row0 + l16) * K + (half << 1);
    const float* __restrict__ wrow = W + (size_t)(half << 1) * M + l16;

    for (int k = 0; k < K; k += 4) {
        v2f a;
        a.x = xrow[k];
        a.y = xrow[k + 1];
        const float* __restrict__ wp = wrow + (size_t)k * M;
#pragma unroll
        for (int t = 0; t < CT; ++t) {
            v2f b;
            b.x = wp[t * 16];
            b.y = wp[t * 16 + M];
            acc[t] = __builtin_amdgcn_wmma_f32_16x16x4_f32(
                false, a, false, b, (short)0, acc[t], false, false);
        }
    }

    float* __restrict__ yrow = Y + (size_t)(row0 + (half << 3)) * M + l16;
#pragma unroll
    for (int t = 0; t < CT; ++t) {
        const float bv = bias[t * 16 + l16];
#pragma unroll
        for (int r = 0; r < 8; ++r)
            yrow[(size_t)r * M + t * 16] = acc[t][r] + bv;
    }
}

// ---------------------------------------------------------------------------
// Edge pass 1: score[e,h] = sum_c leaky_relu(h[src]+h[dst]) * att[h,c]
// and segment-max into menc[dst,h] (ordered-uint atomicMax).
// One wave per edge; lane = channel within a head.
// ---------------------------------------------------------------------------
__global__ __launch_bounds__(256) void edge_score_kernel(
    const float* __restrict__ hmat,
    const long long* __restrict__ src, const long long* __restrict__ dst,
    const float* __restrict__ att,      // [H*C]
    float* __restrict__ score,          // [E*H]
    unsigned* __restrict__ menc,        // [N*H], init 0 (== key below -inf)
    int E) {
    const int e = blockIdx.x * (blockDim.x >> 5) + (threadIdx.x >> 5);
    const int lane = threadIdx.x & 31;
    if (e >= E) return;
    const int s = (int)src[e];
    const int d = (int)dst[e];
    const float* __restrict__ hs = hmat + (size_t)s * GAT_F;
    const float* __restrict__ hd = hmat + (size_t)d * GAT_F;
#pragma unroll
    for (int hh = 0; hh < GAT_H; ++hh) {
        float v = hs[hh * GAT_C + lane] + hd[hh * GAT_C + lane];
        v = (v > 0.0f) ? v : 0.2f * v;                 // leaky_relu, slope 0.2
        float part = v * att[hh * GAT_C + lane];
#pragma unroll
        for (int off = 16; off > 0; off >>= 1)
            part += __shfl_xor(part, off, 32);
        if (lane == 0) {
            score[(size_t)e * GAT_H + hh] = part;
            atomicMax(&menc[(size_t)d * GAT_H + hh], fkey(part));
        }
    }
}

// ---------------------------------------------------------------------------
// Edge pass 2: p = exp(score - m[dst]); denom[dst,h] += p;
// agg[dst,h,c] += p * h[src,h,c].  Normalization deferred to node pass.
// ---------------------------------------------------------------------------
__global__ __launch_bounds__(256) void edge_agg_kernel(
    const float* __restrict__ hmat,
    const long long* __restrict__ src, const long long* __restrict__ dst,
    const float* __restrict__ score,
    const unsigned* __restrict__ menc,
    float* __restrict__ denom,          // [N*H], init 0
    float* __restrict__ agg,            // [N*F], init 0
    int E) {
    const int e = blockIdx.x * (blockDim.x >> 5) + (threadIdx.x >> 5);
    const int lane = threadIdx.x & 31;
    if (e >= E) return;
    const int s = (int)src[e];
    const int d = (int)dst[e];
    const float* __restrict__ hs = hmat + (size_t)s * GAT_F;
    float* __restrict__ ag = agg + (size_t)d * GAT_F;
#pragma unroll
    for (int hh = 0; hh < GAT_H; ++hh) {
        const float m = funkey(menc[(size_t)d * GAT_H + hh]);
        const float p = __expf(score[(size_t)e * GAT_H + hh] - m);
        if (lane == 0)
            atomicAdd(&denom[(size_t)d * GAT_H + hh], p);
        atomicAdd(&ag[hh * GAT_C + lane], p * hs[hh * GAT_C + lane]);
    }
}

// ---------------------------------------------------------------------------
// Node finalize, middle layers (concat heads): x = elu(agg/denom + bias + skip)
// ---------------------------------------------------------------------------
__global__ __launch_bounds__(256) void finalize_mid(
    const float* __restrict__ agg, const float* __restrict__ denom,
    const float* __restrict__ bias, const float* __restrict__ skip,
    float* __restrict__ xnext, int total) {
    const int i = blockIdx.x * blockDim.x + threadIdx.x;
    if (i >= total) return;
    const int f = i & (GAT_F - 1);
    const int n = i >> 7;
    const float dn = fmaxf(denom[(size_t)n * GAT_H + (f >> 5)], 1e-16f);
    float v = agg[i] / dn + bias[f] + skip[i];
    xnext[i] = (v > 0.0f) ? v : (__expf(v) - 1.0f);    // elu, alpha = 1
}

// ---------------------------------------------------------------------------
// Node finalize, last layer (mean over heads, no activation) -> d_out [N x C]
// ---------------------------------------------------------------------------
__global__ __launch_bounds__(256) void finalize_last(
    const float* __restrict__ agg, const float* __restrict__ denom,
    const float* __restrict__ bias, const float* __restrict__ skip,
    float* __restrict__ out, int total) {
    const int i = blockIdx.x * blockDim.x + threadIdx.x;
    if (i >= total) return;
    const int c = i & (GAT_C - 1);
    const int n = i >> 5;
    float acc = 0.0f;
#pragma unroll
    for (int hh = 0; hh < GAT_H; ++hh) {
        const float dn = fmaxf(denom[(size_t)n * GAT_H + hh], 1e-16f);
        acc += agg[(size_t)n * GAT_F + hh * GAT_C + c] / dn;
    }
    out[i] = acc * 0.25f + bias[c] + skip[i];
}

// ---------------------------------------------------------------------------
extern "C" void kernel_launch(void* const* d_in, const int* in_sizes, int n_in,
                              void* d_out, int out_size, void* d_ws, size_t ws_size,
                              hipStream_t stream) {
    (void)n_in; (void)out_size; (void)ws_size;
    const int Nn = GAT_N;
    const int Ee = in_sizes[1] / 2;

    const float*     x0   = (const float*)d_in[0];
    const long long* ei   = (const long long*)d_in[1];   // int64 edge_index (2,E)
    const long long* srcp = ei;
    const long long* dstp = ei + Ee;

    // Workspace layout (bytes). N*F*4 = 25,600,000.
    char* wsb = (char*)d_ws;
    float*    hbuf  = (float*)(wsb + 0);
    float*    aggb  = (float*)(wsb + 25600000);
    float*    skipb = (float*)(wsb + 51200000);
    float*    xcur  = (float*)(wsb + 76800000);          // node features between layers
    float*    score = (float*)(wsb + 102400000);         // E*H*4 = 9,600,000
    unsigned* menc  = (unsigned*)(wsb + 112000000);      // N*H*4 = 800,000
    float*    denom = (float*)(wsb + 112800000);         // N*H*4 = 800,000

    const int rowTiles   = Nn / 16;          // 3125 (50000 % 16 == 0)
    const int gemmBlocks = (rowTiles + 3) / 4;   // 4 waves / block
    const int edgeBlocks = (Ee + 7) / 8;     // wave/edge, 8 waves/block

    for (int l = 0; l < 3; ++l) {
        const float* W    = (const float*)d_in[2 + 6 * l + 0];
        const float* linb = (const float*)d_in[2 + 6 * l + 1];
        const float* att  = (const float*)d_in[2 + 6 * l + 2];
        const float* bias = (const float*)d_in[2 + 6 * l + 3];
        const float* sW   = (const float*)d_in[2 + 6 * l + 4];
        const float* sb   = (const float*)d_in[2 + 6 * l + 5];
        const float* xin  = (l == 0) ? x0 : xcur;
        const int outdim  = (l < 2) ? GAT_F : GAT_C;

        // h = xin @ W + linb   (N x 128)
        gemm_wmma_strip<8><<<gemmBlocks, 128, 0, stream>>>(
            xin, W, linb, hbuf, Nn, GAT_F);

        // skip = xin @ sW + sb (N x outdim)
        if (outdim == GAT_F) {
            gemm_wmma_strip<8><<<gemmBlocks, 128, 0, stream>>>(
                xin, sW, sb, skipb, Nn, GAT_F);
        } else {
            gemm_wmma_strip<2><<<gemmBlocks, 128, 0, stream>>>(
                xin, sW, sb, skipb, Nn, GAT_F);
        }

        hipMemsetAsync(menc,  0, (size_t)Nn * GAT_H * sizeof(unsigned), stream);
        hipMemsetAsync(denom, 0, (size_t)Nn * GAT_H * sizeof(float), stream);
        hipMemsetAsync(aggb,  0, (size_t)Nn * GAT_F * sizeof(float), stream);

        edge_score_kernel<<<edgeBlocks, 256, 0, stream>>>(
            hbuf, srcp, dstp, att, score, menc, Ee);
        edge_agg_kernel<<<edgeBlocks, 256, 0, stream>>>(
            hbuf, srcp, dstp, score, menc, denom, aggb, Ee);

        if (l < 2) {
            const int total = Nn * GAT_F;
            finalize_mid<<<(total + 255) / 256, 256, 0, stream>>>(
                aggb, denom, bias, skipb, xcur, total);
        } else {
            const int total = Nn * GAT_C;
            finalize_last<<<(total + 255) / 256, 256, 0, stream>>>(
                aggb, denom, bias, skipb, (float*)d_out, total);
        }
    }
}